// GIN_dgl_58377195487744
// MI455X (gfx1250) — compile-verified
//
#include <hip/hip_runtime.h>

typedef __attribute__((ext_vector_type(2))) float v2f;
typedef __attribute__((ext_vector_type(8))) float v8f;

#define IN_F  128
#define HID_F 256
#define OUT_F 64

// ---------------------------------------------------------------------------
// Zero-fill (grid-stride)
// ---------------------------------------------------------------------------
__global__ void zero_f32(float* __restrict__ p, long long n) {
  long long i = (long long)blockIdx.x * blockDim.x + threadIdx.x;
  long long stride = (long long)gridDim.x * blockDim.x;
  for (; i < n; i += stride) p[i] = 0.0f;
}

// ---------------------------------------------------------------------------
// Scatter-add: agg[dst[e]][:] += h[src[e]][:]
// One block per edge; blockDim.x == F/4; each lane handles 4 contiguous f32.
// Lowers to fully-coalesced global_load_b128 + 4x global_atomic_add_f32.
// ---------------------------------------------------------------------------
__global__ void scatter_add_f32(const float* __restrict__ h,
                                const long long* __restrict__ src,
                                const long long* __restrict__ dst,
                                float* __restrict__ agg,
                                int F) {
  long long e = blockIdx.x;
  int f = threadIdx.x * 4;
  long long s = src[e];
  long long d = dst[e];
  float4 v = *(const float4*)(h + s * (long long)F + f);
  float* out = agg + d * (long long)F + f;
  atomicAdd(out + 0, v.x);
  atomicAdd(out + 1, v.y);
  atomicAdd(out + 2, v.z);
  atomicAdd(out + 3, v.w);
}

// ---------------------------------------------------------------------------
// C[M,N] = (A1 + A2)[M,K] @ B[K,N] via V_WMMA_F32_16X16X4_F32.
// One wave (32 threads) computes one 16x16 output tile, marching K in steps
// of 4. Fragment layouts per CDNA5 ISA 7.12.2:
//   A (16x4): lane L holds row M=L%16; VGPR0/1 = K = (L/16)*2 + {0,1}
//   B (4x16): lane L holds col N=L%16; VGPR0/1 = K = (L/16)*2 + {0,1}
//   C (16x16): VGPR v, lanes 0-15 -> M=v, lanes 16-31 -> M=v+8; N = L%16
// EXEC is all-ones (full wave, no divergence) as WMMA requires.
// ---------------------------------------------------------------------------
__global__ __launch_bounds__(32)
void gemm_sum_wmma_f32(const float* __restrict__ A1,
                       const float* __restrict__ A2,
                       const float* __restrict__ B,
                       float* __restrict__ C,
                       int K, int N) {
  const int lane = threadIdx.x;
  const int m16  = lane & 15;
  const int kg   = lane >> 4;          // 0 or 1

  const long long rowBase = (long long)blockIdx.x * 16;
  const int       colBase = blockIdx.y * 16;

  const float* a1p = A1 + (rowBase + m16) * (long long)K + kg * 2;
  const float* a2p = A2 + (rowBase + m16) * (long long)K + kg * 2;
  const float* bp  = B  + (long long)(kg * 2) * N + colBase + m16;

  v8f acc = {};
  for (int k0 = 0; k0 < K; k0 += 4) {
    v2f a, b;
    a[0] = a1p[k0 + 0] + a2p[k0 + 0];
    a[1] = a1p[k0 + 1] + a2p[k0 + 1];
    b[0] = bp[(long long)(k0 + 0) * N];
    b[1] = bp[(long long)(k0 + 1) * N];
    acc = __builtin_amdgcn_wmma_f32_16x16x4_f32(
        /*neg_a=*/false, a, /*neg_b=*/false, b,
        /*c_mod=*/(short)0, acc, /*reuse_a=*/false, /*reuse_b=*/false);
  }

  float* cp = C + (rowBase + kg * 8) * (long long)N + colBase + m16;
#pragma unroll
  for (int v = 0; v < 8; ++v) cp[(long long)v * N] = acc[v];
}

// Scalar fallback for remainder rows (unused when M % 16 == 0, as here).
__global__ void gemm_sum_tail(const float* __restrict__ A1,
                              const float* __restrict__ A2,
                              const float* __restrict__ B,
                              float* __restrict__ C,
                              long long rowStart, long long M, int K, int N) {
  long long r = rowStart + blockIdx.x;
  if (r >= M) return;
  for (int n = threadIdx.x; n < N; n += blockDim.x) {
    float s = 0.0f;
    for (int k = 0; k < K; ++k)
      s += (A1[r * K + k] + A2[r * K + k]) * B[(long long)k * N + n];
    C[r * (long long)N + n] = s;
  }
}

// ---------------------------------------------------------------------------
// Orchestration:
//   agg  = d_ws[0 .. N*HID)           (reused for both layers)
//   h1   = d_ws[N*HID .. 2*N*HID)
//   1) agg1 = scatter(x);  h1 = (x + agg1) @ W1
//   2) agg2 = scatter(h1); out = (h1 + agg2) @ W2
// ---------------------------------------------------------------------------
extern "C" void kernel_launch(void* const* d_in, const int* in_sizes, int n_in,
                              void* d_out, int out_size, void* d_ws, size_t ws_size,
                              hipStream_t stream) {
  const float*      x    = (const float*)d_in[0];
  const long long*  esrc = (const long long*)d_in[1];
  const long long*  edst = (const long long*)d_in[2];
  const float*      W1   = (const float*)d_in[3];
  const float*      W2   = (const float*)d_in[4];
  float*            out  = (float*)d_out;

  const long long n_nodes = (long long)in_sizes[0] / IN_F;   // 100000
  const long long n_edges = (long long)in_sizes[1];          // 1600000

  float* agg = (float*)d_ws;
  float* h1  = (float*)d_ws + (size_t)n_nodes * HID_F;

  const long long mTiles = n_nodes / 16;
  const long long mTail  = n_nodes % 16;

  // ---- Layer 1 ----
  zero_f32<<<2048, 256, 0, stream>>>(agg, n_nodes * IN_F);
  scatter_add_f32<<<(unsigned)n_edges, IN_F / 4, 0, stream>>>(x, esrc, edst, agg, IN_F);
  if (mTiles > 0)
    gemm_sum_wmma_f32<<<dim3((unsigned)mTiles, HID_F / 16), 32, 0, stream>>>(
        x, agg, W1, h1, IN_F, HID_F);
  if (mTail > 0)
    gemm_sum_tail<<<(unsigned)mTail, 64, 0, stream>>>(
        x, agg, W1, h1, mTiles * 16, n_nodes, IN_F, HID_F);

  // ---- Layer 2 ----
  zero_f32<<<2048, 256, 0, stream>>>(agg, n_nodes * HID_F);
  scatter_add_f32<<<(unsigned)n_edges, HID_F / 4, 0, stream>>>(h1, esrc, edst, agg, HID_F);
  if (mTiles > 0)
    gemm_sum_wmma_f32<<<dim3((unsigned)mTiles, OUT_F / 16), 32, 0, stream>>>(
        h1, agg, W2, out, HID_F, OUT_F);
  if (mTail > 0)
    gemm_sum_tail<<<(unsigned)mTail, 64, 0, stream>>>(
        h1, agg, W2, out, mTiles * 16, n_nodes, HID_F, OUT_F);
}